// PNA_10187662426377
// MI455X (gfx1250) — compile-verified
//
#include <hip/hip_runtime.h>
#include <math.h>

// ---------------- constants ----------------
#define N_NODES 30000
#define N_EDGES 300000
#define N_GRAPHS 1000
#define IN_DIM 9
#define HIDDEN 64
#define AVG_DEG_LOG 2.3978952727983707f   // log(11.0)

typedef float v2f_t __attribute__((ext_vector_type(2)));
typedef float v8f_t __attribute__((ext_vector_type(8)));

__device__ __forceinline__ v8f_t wmma_f32_k4(v2f_t a, v2f_t b, v8f_t c) {
    // D(16x16,f32) += A(16x4,f32) * B(4x16,f32)
    return __builtin_amdgcn_wmma_f32_16x16x4_f32(false, a, false, b, (short)0, c, false, false);
}

// order-preserving float <-> uint key (for atomicMin/Max on floats)
__device__ __forceinline__ unsigned fkey(float f) {
    unsigned b = __float_as_uint(f);
    unsigned m = ((unsigned)((int)b >> 31)) | 0x80000000u;
    return b ^ m;
}
__device__ __forceinline__ float fdec(unsigned u) {
    unsigned m = ((int)u >= 0) ? 0xFFFFFFFFu : 0x80000000u;
    return __uint_as_float(u ^ m);
}

// ---------------- init kernels ----------------
__global__ void fillU(unsigned* p, unsigned v, int n) {
    int i = blockIdx.x * blockDim.x + threadIdx.x;
    if (i < n) p[i] = v;
}
__global__ void fillF(float* p, float v, int n) {
    int i = blockIdx.x * blockDim.x + threadIdx.x;
    if (i < n) p[i] = v;
}
__global__ void countK(const int* __restrict__ dst, unsigned* cnt, int E) {
    int e = blockIdx.x * blockDim.x + threadIdx.x;
    if (e < E) atomicAdd(&cnt[dst[e]], 1u);
}

// ---------------- layer 0 (f_in = 9), plain VALU ----------------
__global__ void edge_l0(const float* __restrict__ x0, const float* __restrict__ eattr,
                        const int* __restrict__ src, const int* __restrict__ dst,
                        const float* __restrict__ We, const float* __restrict__ be,
                        const float* __restrict__ Wpre, const float* __restrict__ bpre,
                        float* sum, float* sumsq, unsigned* mnU, unsigned* mxU, int E) {
    int e = blockIdx.x * blockDim.x + threadIdx.x;
    if (e >= E) return;
    int s = src[e], d = dst[e];
    float h[27];
#pragma unroll
    for (int c = 0; c < 9; c++) { h[c] = x0[d * 9 + c]; h[9 + c] = x0[s * 9 + c]; }
    float ea0 = eattr[e * 3], ea1 = eattr[e * 3 + 1], ea2 = eattr[e * 3 + 2];
#pragma unroll
    for (int c = 0; c < 9; c++)
        h[18 + c] = be[c] + ea0 * We[c] + ea1 * We[9 + c] + ea2 * We[18 + c];
    for (int t = 0; t < 4; t++) {
        for (int g = 0; g < 9; g++) {
            float acc = bpre[t * 9 + g];
#pragma unroll
            for (int k = 0; k < 27; k++) acc += h[k] * Wpre[t * 243 + k * 9 + g];
            int fc = t * 9 + g;
            atomicAdd(&sum[d * 36 + fc], acc);
            atomicAdd(&sumsq[d * 36 + fc], acc * acc);
            unsigned u = fkey(acc);
            atomicMin(&mnU[d * 36 + fc], u);
            atomicMax(&mxU[d * 36 + fc], u);
        }
    }
}

__global__ void post_l0(const float* __restrict__ x0, const unsigned* __restrict__ cnt,
                        const float* __restrict__ sum, const float* __restrict__ sumsq,
                        const unsigned* __restrict__ mnU, const unsigned* __restrict__ mxU,
                        const float* __restrict__ Wpost, const float* __restrict__ bpost,
                        const float* __restrict__ Wlin, const float* __restrict__ blin,
                        const float* __restrict__ gam, const float* __restrict__ bet,
                        float* __restrict__ xout, int N) {
    int n = blockIdx.x * blockDim.x + threadIdx.x;
    if (n >= N) return;
    unsigned c = cnt[n];
    float c1 = (c > 0) ? (float)c : 1.0f;
    float lg = logf(c1 + 1.0f);
    float amp = lg / AVG_DEG_LOG, att = AVG_DEG_LOG / lg;
    float h64[64];
    for (int t = 0; t < 4; t++) {
        float aggv[36];
#pragma unroll
        for (int j = 0; j < 9; j++) {
            int f = t * 9 + j;
            float mean = sum[n * 36 + f] / c1;
            float m2 = sumsq[n * 36 + f] / c1;
            aggv[j] = mean;
            aggv[9 + j]  = (c > 0) ? fdec(mnU[n * 36 + f]) : 0.0f;
            aggv[18 + j] = (c > 0) ? fdec(mxU[n * 36 + f]) : 0.0f;
            aggv[27 + j] = sqrtf(fmaxf(m2 - mean * mean, 0.0f) + 1e-5f);
        }
        const float* Wp = Wpost + t * (117 * 16);
        for (int g = 0; g < 16; g++) {
            float acc = bpost[t * 16 + g];
#pragma unroll
            for (int c9 = 0; c9 < 9; c9++) acc += x0[n * 9 + c9] * Wp[c9 * 16 + g];
            for (int j = 0; j < 36; j++) {
                float a = aggv[j];
                acc += a * Wp[(9 + j) * 16 + g];
                acc += a * amp * Wp[(45 + j) * 16 + g];
                acc += a * att * Wp[(81 + j) * 16 + g];
            }
            h64[t * 16 + g] = acc;
        }
    }
    float t1[64];
    float mu = 0.0f;
    for (int m = 0; m < 64; m++) {
        float acc = blin[m];
        for (int k = 0; k < 64; k++) acc += h64[k] * Wlin[k * 64 + m];
        t1[m] = acc; mu += acc;
    }
    mu *= (1.0f / 64.0f);
    float var = 0.0f;
    for (int m = 0; m < 64; m++) { float d = t1[m] - mu; var += d * d; }
    var *= (1.0f / 64.0f);
    float rs = rsqrtf(var + 1e-5f);
    for (int m = 0; m < 64; m++) {
        float v = (t1[m] - mu) * rs * gam[m] + bet[m];
        xout[n * 64 + m] = fmaxf(v, 0.0f);
    }
}

// ---------------- layers 1,2 edge pre-MLP: WMMA f32 16x16x4 ----------------
// One wave (32 threads) per 16-edge tile. h = [x_dst | x_src | e] : [16 x 192]
// m[e, t*64+g] = h @ Wpre[t] + bpre[t]; fused scatter aggregation via atomics.
__global__ __launch_bounds__(32) void edge_wmma(
        const float* __restrict__ x, const float* __restrict__ eattr,
        const int* __restrict__ src, const int* __restrict__ dst,
        const float* __restrict__ We, const float* __restrict__ be,
        const float* __restrict__ Wpre, const float* __restrict__ bpre,
        float* sum, float* sumsq, unsigned* mnU, unsigned* mxU) {
    __shared__ float hA[16 * 192];
    __shared__ float hB[192 * 16];
    __shared__ int sDst[16], sSrc[16];
    int lane = threadIdx.x;
    int lo = lane & 15, hi = lane >> 4;
    int eb = blockIdx.x * 16;
    if (lane < 16) { sDst[lane] = dst[eb + lane]; sSrc[lane] = src[eb + lane]; }
    __syncthreads();
    for (int idx = lane; idx < 16 * 192; idx += 32) {
        int row = idx / 192, col = idx - row * 192;
        float v;
        if (col < 64) v = x[sDst[row] * 64 + col];
        else if (col < 128) v = x[sSrc[row] * 64 + (col - 64)];
        else {
            int cc = col - 128;
            const float* ea = &eattr[(eb + row) * 3];
            v = be[cc] + ea[0] * We[cc] + ea[1] * We[64 + cc] + ea[2] * We[128 + cc];
        }
        hA[idx] = v;
    }
    __syncthreads();
    for (int nt = 0; nt < 16; nt++) {
        int tower = nt >> 2, sub = nt & 3;
        int gbase = sub * 16;
        const float* Wt = Wpre + tower * (192 * 64);
        for (int idx = lane; idx < 192 * 16; idx += 32) {
            int k = idx >> 4, gg = idx & 15;
            hB[idx] = Wt[k * 64 + gbase + gg];
        }
        __syncthreads();
        v8f_t acc = {0.f, 0.f, 0.f, 0.f, 0.f, 0.f, 0.f, 0.f};
        for (int k0 = 0; k0 < 192; k0 += 4) {
            int k = k0 + 2 * hi;
            v2f_t a; a.x = hA[lo * 192 + k]; a.y = hA[lo * 192 + k + 1];
            v2f_t b; b.x = hB[k * 16 + lo];  b.y = hB[(k + 1) * 16 + lo];
            acc = wmma_f32_k4(a, b, acc);
        }
        int fc = tower * 64 + gbase + lo;         // feature col within [0,256)
        float bias = bpre[fc];
#pragma unroll
        for (int r = 0; r < 8; r++) {
            int node = sDst[r + 8 * hi];
            float v = acc[r] + bias;
            atomicAdd(&sum[node * 256 + fc], v);
            atomicAdd(&sumsq[node * 256 + fc], v * v);
            unsigned u = fkey(v);
            atomicMin(&mnU[node * 256 + fc], u);
            atomicMax(&mxU[node * 256 + fc], u);
        }
        __syncthreads();
    }
}

// ---------------- layers 1,2 node post-MLP: WMMA f32 16x16x4 ----------------
// One wave per 16-node tile: build [16 x 832] per tower, Wpost -> [16x16],
// concat towers -> [16x64], Wlin -> [16x64], layernorm + relu -> x_out.
__global__ __launch_bounds__(32) void node_post_wmma(
        const float* __restrict__ x, const unsigned* __restrict__ cnt,
        const float* __restrict__ sum, const float* __restrict__ sumsq,
        const unsigned* __restrict__ mnU, const unsigned* __restrict__ mxU,
        const float* __restrict__ Wpost, const float* __restrict__ bpost,
        const float* __restrict__ Wlin, const float* __restrict__ blin,
        const float* __restrict__ gam, const float* __restrict__ bet,
        float* __restrict__ xout) {
    __shared__ float sA[16 * 832];
    __shared__ float sH[16 * 64];
    __shared__ float sG[16 * 64];
    __shared__ float sC1[16], sAmp[16], sAtt[16], sMu[16], sRs[16];
    __shared__ unsigned sHas[16];
    int lane = threadIdx.x;
    int lo = lane & 15, hi = lane >> 4;
    int nb = blockIdx.x * 16;
    if (lane < 16) {
        unsigned c = cnt[nb + lane];
        float c1 = (c > 0) ? (float)c : 1.0f;
        sC1[lane] = c1; sHas[lane] = c;
        float lg = logf(c1 + 1.0f);
        sAmp[lane] = lg / AVG_DEG_LOG;
        sAtt[lane] = AVG_DEG_LOG / lg;
    }
    __syncthreads();
    for (int t = 0; t < 4; t++) {
        for (int idx = lane; idx < 16 * 832; idx += 32) {
            int row = idx / 832, col = idx - row * 832;
            int node = nb + row;
            float v;
            if (col < 64) v = x[node * 64 + col];
            else {
                int cc = col - 64;
                int s = cc >> 8;       // 0: agg, 1: agg*amp, 2: agg*att
                int j = cc & 255;
                int kind = j >> 6;     // 0: mean 1: min 2: max 3: std
                int f = t * 64 + (j & 63);
                float c1 = sC1[row];
                float v0;
                if (kind == 0) v0 = sum[node * 256 + f] / c1;
                else if (kind == 1) v0 = sHas[row] ? fdec(mnU[node * 256 + f]) : 0.0f;
                else if (kind == 2) v0 = sHas[row] ? fdec(mxU[node * 256 + f]) : 0.0f;
                else {
                    float mean = sum[node * 256 + f] / c1;
                    float m2 = sumsq[node * 256 + f] / c1;
                    v0 = sqrtf(fmaxf(m2 - mean * mean, 0.0f) + 1e-5f);
                }
                float sc = (s == 0) ? 1.0f : ((s == 1) ? sAmp[row] : sAtt[row]);
                v = v0 * sc;
            }
            sA[idx] = v;
        }
        __syncthreads();
        const float* Wp = Wpost + t * (832 * 16);
        v8f_t acc = {0.f, 0.f, 0.f, 0.f, 0.f, 0.f, 0.f, 0.f};
        for (int k0 = 0; k0 < 832; k0 += 4) {
            int k = k0 + 2 * hi;
            v2f_t a; a.x = sA[lo * 832 + k]; a.y = sA[lo * 832 + k + 1];
            v2f_t b; b.x = Wp[k * 16 + lo];  b.y = Wp[(k + 1) * 16 + lo];
            acc = wmma_f32_k4(a, b, acc);
        }
        float bb = bpost[t * 16 + lo];
#pragma unroll
        for (int r = 0; r < 8; r++)
            sH[(r + 8 * hi) * 64 + t * 16 + lo] = acc[r] + bb;
        __syncthreads();
    }
    // Wlin: [16x64] @ [64x64]
    for (int nt = 0; nt < 4; nt++) {
        v8f_t acc = {0.f, 0.f, 0.f, 0.f, 0.f, 0.f, 0.f, 0.f};
        for (int k0 = 0; k0 < 64; k0 += 4) {
            int k = k0 + 2 * hi;
            v2f_t a; a.x = sH[lo * 64 + k];           a.y = sH[lo * 64 + k + 1];
            v2f_t b; b.x = Wlin[k * 64 + nt * 16 + lo]; b.y = Wlin[(k + 1) * 64 + nt * 16 + lo];
            acc = wmma_f32_k4(a, b, acc);
        }
        float bb = blin[nt * 16 + lo];
#pragma unroll
        for (int r = 0; r < 8; r++)
            sG[(r + 8 * hi) * 64 + nt * 16 + lo] = acc[r] + bb;
    }
    __syncthreads();
    if (lane < 16) {
        float mu = 0.0f;
        for (int cc = 0; cc < 64; cc++) mu += sG[lane * 64 + cc];
        mu *= (1.0f / 64.0f);
        float var = 0.0f;
        for (int cc = 0; cc < 64; cc++) { float d = sG[lane * 64 + cc] - mu; var += d * d; }
        var *= (1.0f / 64.0f);
        sMu[lane] = mu;
        sRs[lane] = rsqrtf(var + 1e-5f);
    }
    __syncthreads();
    for (int idx = lane; idx < 16 * 64; idx += 32) {
        int row = idx >> 6, cc = idx & 63;
        float v = (sG[idx] - sMu[row]) * sRs[row] * gam[cc] + bet[cc];
        xout[(nb + row) * 64 + cc] = fmaxf(v, 0.0f);
    }
}

// ---------------- pooling + graph MLP head ----------------
__global__ void poolK(const float* __restrict__ h, const int* __restrict__ batch,
                      float* pooled, int total) {
    int i = blockIdx.x * blockDim.x + threadIdx.x;
    if (i >= total) return;
    int n = i >> 6, c = i & 63;
    atomicAdd(&pooled[batch[n] * 64 + c], h[i]);
}

__device__ __forceinline__ void dense64_ln_relu(float* h, const float* W, const float* b,
                                                const float* gam, const float* bet) {
    float t[64];
    float mu = 0.0f;
    for (int n = 0; n < 64; n++) {
        float acc = b[n];
        for (int k = 0; k < 64; k++) acc += h[k] * W[k * 64 + n];
        t[n] = acc; mu += acc;
    }
    mu *= (1.0f / 64.0f);
    float var = 0.0f;
    for (int n = 0; n < 64; n++) { float d = t[n] - mu; var += d * d; }
    var *= (1.0f / 64.0f);
    float rs = rsqrtf(var + 1e-5f);
    for (int n = 0; n < 64; n++) {
        float v = (t[n] - mu) * rs * gam[n] + bet[n];
        h[n] = fmaxf(v, 0.0f);
    }
}

__global__ void mlpK(const float* __restrict__ pooled,
                     const float* W1, const float* b1, const float* g1, const float* bt1,
                     const float* W2, const float* b2, const float* g2, const float* bt2,
                     const float* Wo1, const float* bo1,
                     const float* Wo2, const float* bo2,
                     const float* Wo3, const float* bo3,
                     float* __restrict__ out, int G) {
    int g = blockIdx.x * blockDim.x + threadIdx.x;
    if (g >= G) return;
    float h[64];
    for (int k = 0; k < 64; k++) h[k] = pooled[g * 64 + k];
    dense64_ln_relu(h, W1, b1, g1, bt1);
    dense64_ln_relu(h, W2, b2, g2, bt2);
    float o1[32];
    for (int n = 0; n < 32; n++) {
        float acc = bo1[n];
        for (int k = 0; k < 64; k++) acc += h[k] * Wo1[k * 32 + n];
        o1[n] = fmaxf(acc, 0.0f);
    }
    float o2[16];
    for (int n = 0; n < 16; n++) {
        float acc = bo2[n];
        for (int k = 0; k < 32; k++) acc += o1[k] * Wo2[k * 16 + n];
        o2[n] = fmaxf(acc, 0.0f);
    }
    for (int n = 0; n < 3; n++) {
        float acc = bo3[n];
        for (int k = 0; k < 16; k++) acc += o2[k] * Wo3[k * 3 + n];
        out[g * 3 + n] = acc * acc + 1e-6f;
    }
}

// ---------------- host launch ----------------
extern "C" void kernel_launch(void* const* d_in, const int* in_sizes, int n_in,
                              void* d_out, int out_size, void* d_ws, size_t ws_size,
                              hipStream_t stream) {
    (void)in_sizes; (void)n_in; (void)out_size; (void)ws_size;
    const int E = N_EDGES, N = N_NODES, G = N_GRAPHS;

    const float* X0   = (const float*)d_in[0];
    const float* EATT = (const float*)d_in[1];
    const int*   EI   = (const int*)d_in[2];
    const int*   SRC  = EI;
    const int*   DST  = EI + E;
    const int*   BAT  = (const int*)d_in[3];
    // conv params, sorted keys: We,Wlin,Wpost,Wpre,be,blin,bpost,bpre,bt,g
#define CP(l, k) ((const float*)d_in[4 + (l) * 10 + (k)])
    const float* mW1  = (const float*)d_in[34];
    const float* mW2  = (const float*)d_in[35];
    const float* mb1  = (const float*)d_in[36];
    const float* mb2  = (const float*)d_in[37];
    const float* mbt1 = (const float*)d_in[38];
    const float* mbt2 = (const float*)d_in[39];
    const float* mg1  = (const float*)d_in[40];
    const float* mg2  = (const float*)d_in[41];
    const float* oW1  = (const float*)d_in[42];
    const float* oW2  = (const float*)d_in[43];
    const float* oW3  = (const float*)d_in[44];
    const float* ob1  = (const float*)d_in[45];
    const float* ob2  = (const float*)d_in[46];
    const float* ob3  = (const float*)d_in[47];

    char* base = (char*)d_ws;
    size_t off = 0;
    auto carve = [&](size_t bytes) -> char* {
        char* p = base + off;
        off = (off + bytes + 255) & ~(size_t)255;
        return p;
    };
    float*    xA     = (float*)carve((size_t)N * 64 * 4);
    float*    xB     = (float*)carve((size_t)N * 64 * 4);
    float*    sum    = (float*)carve((size_t)N * 256 * 4);
    float*    sumsq  = (float*)carve((size_t)N * 256 * 4);
    unsigned* mnU    = (unsigned*)carve((size_t)N * 256 * 4);
    unsigned* mxU    = (unsigned*)carve((size_t)N * 256 * 4);
    unsigned* cnt    = (unsigned*)carve((size_t)N * 4);
    float*    pooled = (float*)carve((size_t)G * 64 * 4);

    const int TB = 256;
    auto blocks = [](int n, int tb) { return (n + tb - 1) / tb; };

    // in-degree counts (reused by all layers)
    fillU<<<blocks(N, TB), TB, 0, stream>>>(cnt, 0u, N);
    countK<<<blocks(E, TB), TB, 0, stream>>>(DST, cnt, E);

    // ---- layer 0 (f_in = 9) ----
    {
        int n36 = N * 36;
        fillF<<<blocks(n36, TB), TB, 0, stream>>>(sum, 0.0f, n36);
        fillF<<<blocks(n36, TB), TB, 0, stream>>>(sumsq, 0.0f, n36);
        fillU<<<blocks(n36, TB), TB, 0, stream>>>(mnU, 0xFFFFFFFFu, n36);
        fillU<<<blocks(n36, TB), TB, 0, stream>>>(mxU, 0u, n36);
        edge_l0<<<blocks(E, TB), TB, 0, stream>>>(X0, EATT, SRC, DST,
            CP(0, 0), CP(0, 4), CP(0, 3), CP(0, 7), sum, sumsq, mnU, mxU, E);
        post_l0<<<blocks(N, TB), TB, 0, stream>>>(X0, cnt, sum, sumsq, mnU, mxU,
            CP(0, 2), CP(0, 6), CP(0, 1), CP(0, 5), CP(0, 9), CP(0, 8), xA, N);
    }

    // ---- layers 1,2 (f_in = 64, WMMA) ----
    const float* xin = xA;
    float* xout = xB;
    for (int l = 1; l <= 2; l++) {
        int n256 = N * 256;
        fillF<<<blocks(n256, TB), TB, 0, stream>>>(sum, 0.0f, n256);
        fillF<<<blocks(n256, TB), TB, 0, stream>>>(sumsq, 0.0f, n256);
        fillU<<<blocks(n256, TB), TB, 0, stream>>>(mnU, 0xFFFFFFFFu, n256);
        fillU<<<blocks(n256, TB), TB, 0, stream>>>(mxU, 0u, n256);
        edge_wmma<<<E / 16, 32, 0, stream>>>(xin, EATT, SRC, DST,
            CP(l, 0), CP(l, 4), CP(l, 3), CP(l, 7), sum, sumsq, mnU, mxU);
        node_post_wmma<<<N / 16, 32, 0, stream>>>(xin, cnt, sum, sumsq, mnU, mxU,
            CP(l, 2), CP(l, 6), CP(l, 1), CP(l, 5), CP(l, 9), CP(l, 8), xout);
        const float* tmp = xin;
        xin = xout;
        xout = (float*)tmp;
    }

    // ---- pooling + head ----
    fillF<<<blocks(G * 64, TB), TB, 0, stream>>>(pooled, 0.0f, G * 64);
    poolK<<<blocks(N * 64, TB), TB, 0, stream>>>(xin, BAT, pooled, N * 64);
    mlpK<<<blocks(G, TB), TB, 0, stream>>>(pooled,
        mW1, mb1, mg1, mbt1, mW2, mb2, mg2, mbt2,
        oW1, ob1, oW2, ob2, oW3, ob3, (float*)d_out, G);
#undef CP
}